// FeatureFusionModule_38113539785018
// MI455X (gfx1250) — compile-verified
//
#include <hip/hip_runtime.h>
#include <stdint.h>

// ---------------------------------------------------------------------------
// FeatureFusion: out[c, :] = sum_{v=0..3} concat(shared, special)[4c+v, :]
//                pout[c]   = pids[4c]
// Memory-bound (~640 MB/launch -> ~27us floor @ 23.3 TB/s). No matmul content,
// so no WMMA; instead use the CDNA5 Tensor Data Mover to stage each 4x1024
// f32 tile into LDS (TENSORcnt-tracked), then b128 LDS reads + VALU adds +
// non-temporal b128 global stores.
// ---------------------------------------------------------------------------

typedef __attribute__((ext_vector_type(4))) float    f32x4;
typedef __attribute__((ext_vector_type(4))) unsigned u32x4;
typedef __attribute__((ext_vector_type(8))) int      i32x8;
typedef __attribute__((ext_vector_type(4))) int      i32x4;

#define VIEWS   4
#define D_FIXED 1024   // feature dim of each input (reference: 1024)

// One block per fused output row c. 256 threads = 8 waves.
// LDS: [0 .. 4095]  = shared rows 4c..4c+3   (16 KB)
//      [4096..8191] = special rows 4c..4c+3  (16 KB)
__global__ __launch_bounds__(256) void fusion_tdm_kernel(
    const float* __restrict__ A,      // shared_features  [bs, 1024]
    const float* __restrict__ B,      // special_features [bs, 1024]
    const int*   __restrict__ pids,   // [bs]
    float*       __restrict__ out,    // [bs/4, 2048]
    int*         __restrict__ pout)   // [bs/4]
{
    __shared__ float smem[2 * VIEWS * D_FIXED];   // 32 KB
    const int c = blockIdx.x;
    const int t = threadIdx.x;
    const int Q = D_FIXED / 4;                    // 256 float4 per input row

#if defined(__HIP_DEVICE_COMPILE__) && \
    __has_builtin(__builtin_amdgcn_tensor_load_to_lds) && \
    __has_builtin(__builtin_amdgcn_s_wait_tensorcnt)
    // ---- CDNA5 TDM path: wave 0 issues two tensor DMA loads ----
    if (t < 32) {
        const unsigned ldsA = (unsigned)(uintptr_t)(void*)&smem[0];
        const unsigned ldsB = (unsigned)(uintptr_t)(void*)&smem[VIEWS * D_FIXED];
        const unsigned long long gA =
            (unsigned long long)(uintptr_t)(A + (size_t)c * VIEWS * D_FIXED);
        const unsigned long long gB =
            (unsigned long long)(uintptr_t)(B + (size_t)c * VIEWS * D_FIXED);

        // D# group 1 (256b): workgroup_mask=0 (not in cluster), data_size=4B,
        // tensor_dim0=1024, tensor_dim1=65536, tile_dim0=1024, tile_dim1=4,
        // tile_dim2=0, tensor_dim0_stride=1024.
        i32x8 g1;
        g1[0] = 0x20000;                                  // data_size=2 (4B)
        g1[1] = (int)((unsigned)D_FIXED << 16);           // tensor_dim0 lo16
        g1[2] = 0;                                        // dim0 hi16 | dim1 lo16
        g1[3] = (int)(((unsigned)D_FIXED << 16) | 1u);    // tile_dim0 | dim1 hi16 (65536)
        g1[4] = VIEWS;                                    // tile_dim1=4, tile_dim2=0
        g1[5] = D_FIXED;                                  // tensor_dim0_stride lo32
        g1[6] = 0;
        g1[7] = 0;
        // Groups 2/3: dims 2..4 unused; keep tensor_dim2/3/4 = 1 so the
        // (single) z/w iteration is in-bounds, tile_dim3/4 = 0.
        i32x4 g2 = {1, 1, 0, 0};                          // tensor_dim2=1, tensor_dim3=1
        i32x4 g3 = {0, 1 << 16, 0, 0};                    // tensor_dim4=1

        // D# group 0 (128b): count=1, lds_addr, global_addr(57b), type=2.
        u32x4 g0a, g0b;
        g0a[0] = 1u;
        g0a[1] = ldsA;
        g0a[2] = (unsigned)gA;
        g0a[3] = (unsigned)((gA >> 32) & 0x01FFFFFFull) | 0x80000000u;
        g0b[0] = 1u;
        g0b[1] = ldsB;
        g0b[2] = (unsigned)gB;
        g0b[3] = (unsigned)((gB >> 32) & 0x01FFFFFFull) | 0x80000000u;

#if __clang_major__ >= 23
        i32x8 gpad = {};   // clang-23 / therock headers: 6-arg form
        __builtin_amdgcn_tensor_load_to_lds(g0a, g1, g2, g3, gpad, 0);
        __builtin_amdgcn_tensor_load_to_lds(g0b, g1, g2, g3, gpad, 0);
#else
        __builtin_amdgcn_tensor_load_to_lds(g0a, g1, g2, g3, 0);      // ROCm 7.2 5-arg
        __builtin_amdgcn_tensor_load_to_lds(g0b, g1, g2, g3, 0);
#endif
        __builtin_amdgcn_s_wait_tensorcnt(0);
    }
    __syncthreads();

    const f32x4* a4 = (const f32x4*)&smem[0];
    const f32x4* b4 = (const f32x4*)&smem[VIEWS * D_FIXED];
#else
    // ---- Fallback (builtin unavailable): direct b128 global loads ----
    const f32x4* a4 = (const f32x4*)(A + (size_t)c * VIEWS * D_FIXED);
    const f32x4* b4 = (const f32x4*)(B + (size_t)c * VIEWS * D_FIXED);
#endif

    // Each thread produces 2 output float4s (one per input half).
    f32x4 sA = a4[t] + a4[Q + t] + a4[2 * Q + t] + a4[3 * Q + t];
    f32x4 sB = b4[t] + b4[Q + t] + b4[2 * Q + t] + b4[3 * Q + t];

    f32x4* o4 = (f32x4*)(out + (size_t)c * 2 * D_FIXED);
    __builtin_nontemporal_store(sA, o4 + t);        // streamed once, bypass caches
    __builtin_nontemporal_store(sB, o4 + Q + t);

    if (t == 0) pout[c] = pids[(size_t)c * VIEWS];
}

// Generic fallback for d != 1024 (not expected with this reference).
__global__ __launch_bounds__(256) void fusion_plain_kernel(
    const float* __restrict__ A, const float* __restrict__ B,
    const int* __restrict__ pids, float* __restrict__ out,
    int* __restrict__ pout, int chunk, int d)
{
    const int q4 = d / 4;
    const long long total4 = (long long)chunk * (2 * q4);
    const long long stride = (long long)gridDim.x * blockDim.x;
    for (long long i = (long long)blockIdx.x * blockDim.x + threadIdx.x;
         i < total4; i += stride) {
        long long c = i / (2 * q4);
        int q = (int)(i - c * (2 * q4));
        const f32x4* src = (const f32x4*)((q < q4) ? A : B);
        int col = (q < q4) ? q : q - q4;
        const f32x4* base = src + (size_t)c * VIEWS * q4 + col;
        f32x4 s = base[0] + base[q4] + base[2 * q4] + base[3 * q4];
        __builtin_nontemporal_store(s, (f32x4*)out + i);
    }
    for (long long i = (long long)blockIdx.x * blockDim.x + threadIdx.x;
         i < chunk; i += stride)
        pout[i] = pids[i * VIEWS];
}

extern "C" void kernel_launch(void* const* d_in, const int* in_sizes, int n_in,
                              void* d_out, int out_size, void* d_ws, size_t ws_size,
                              hipStream_t stream) {
    const float* A    = (const float*)d_in[0];
    const float* B    = (const float*)d_in[1];
    const int*   pids = (const int*)d_in[2];

    const int bs    = in_sizes[2];          // 65536
    const int d     = in_sizes[0] / bs;     // 1024
    const int chunk = bs / VIEWS;           // 16384

    float* out  = (float*)d_out;
    int*   pout = (int*)((float*)d_out + (size_t)chunk * 2 * d);

    if (d == D_FIXED) {
        fusion_tdm_kernel<<<chunk, 256, 0, stream>>>(A, B, pids, out, pout);
    } else {
        fusion_plain_kernel<<<4096, 256, 0, stream>>>(A, B, pids, out, pout, chunk, d);
    }
}